// GNN_74191265071300
// MI455X (gfx1250) — compile-verified
//
#include <hip/hip_runtime.h>
#include <hip/hip_bf16.h>

#define N_NODES  50000
#define N_EDGES  800000
#define FEAT     64
#define N_GRAPHS 500
#define NF       (N_NODES * FEAT)

typedef __attribute__((ext_vector_type(2))) float v2f;
typedef __attribute__((ext_vector_type(8))) float v8f;

// ---------------------------------------------------------------- zero fill
__global__ void gnn_zero_kernel(float* __restrict__ p, int n) {
    int i = blockIdx.x * blockDim.x + threadIdx.x;
    if (i < n) p[i] = 0.0f;
}

// ------------------------------------------------- edge gather + scatter-add
// one thread per (edge, 4-feature group): float4 gather from h[src],
// 4 atomic f32 adds into agg[dst]
__global__ __launch_bounds__(256) void gnn_edge_scatter_kernel(
    const int* __restrict__ ei,       // [2, E] (src row then dst row)
    const float* __restrict__ h,      // [N, 64]
    float* __restrict__ agg)          // [N, 64]
{
    int idx = blockIdx.x * blockDim.x + threadIdx.x;
    int e  = idx >> 4;
    int gp = (idx & 15) << 2;
    if (e >= N_EDGES) return;
    int src = ei[e];
    int dst = ei[N_EDGES + e];
    const float4 v = *(const float4*)(h + src * FEAT + gp);
    float* a = agg + dst * FEAT + gp;
    atomicAdd(a + 0, v.x);
    atomicAdd(a + 1, v.y);
    atomicAdd(a + 2, v.z);
    atomicAdd(a + 3, v.w);
}

// ------------------------------------------- fused dual-GEMM + bias + ReLU
// OUT = relu(H @ Wroot + AGG @ Wrel + b), all fp32 via V_WMMA_F32_16X16X4_F32.
// One wave -> one 16-row x 64-col output strip. Weights staged in LDS.
__global__ __launch_bounds__(128) void gnn_gemm_relu_kernel(
    const float* __restrict__ H,      // [N, 64]
    const float* __restrict__ AGG,    // [N, 64]
    const float* __restrict__ Wroot,  // [64, 64] (k-major: W[k*64+n])
    const float* __restrict__ Wrel,   // [64, 64]
    const float* __restrict__ bias,   // [64]
    float* __restrict__ OUT,          // [N, 64]
    int nrows)
{
    __shared__ float sWr[FEAT * FEAT]; // Wroot
    __shared__ float sWl[FEAT * FEAT]; // Wrel

    for (int i = threadIdx.x; i < FEAT * FEAT; i += 128) {
        sWr[i] = Wroot[i];
        sWl[i] = Wrel[i];
    }
    __syncthreads();

    const int wave  = threadIdx.x >> 5;
    const int lane  = threadIdx.x & 31;
    const int tile  = blockIdx.x * 4 + wave;      // 16-row tile id
    const int rowb  = tile * 16;
    if (rowb >= nrows) return;                    // wave-uniform exit

    const int m     = lane & 15;                  // row (A) / col (B,C,D)
    const int khalf = lane >> 4;                  // 0 or 1

    // accumulators: 4 col-tiles of 16x16, init with broadcast bias (per col)
    v8f acc[4];
#pragma unroll
    for (int t = 0; t < 4; ++t) {
        float bv = bias[t * 16 + m];
#pragma unroll
        for (int v = 0; v < 8; ++v) acc[t][v] = bv;
    }

    const float* hrow = H   + (rowb + m) * FEAT;
    const float* arow = AGG + (rowb + m) * FEAT;

#pragma unroll
    for (int k = 0; k < FEAT; k += 4) {
        const int ka = k + 2 * khalf;             // even -> 8B aligned
        // A fragments (16x4): lane holds A[m][ka], A[m][ka+1]
        v2f aH = *(const v2f*)(hrow + ka);
        v2f aA = *(const v2f*)(arow + ka);
#pragma unroll
        for (int t = 0; t < 4; ++t) {
            const int col = t * 16 + m;
            // B fragments (4x16): lane holds B[ka][col], B[ka+1][col]
            v2f bR; bR.x = sWr[ka * FEAT + col]; bR.y = sWr[(ka + 1) * FEAT + col];
            v2f bL; bL.x = sWl[ka * FEAT + col]; bL.y = sWl[(ka + 1) * FEAT + col];
            acc[t] = __builtin_amdgcn_wmma_f32_16x16x4_f32(
                false, aH, false, bR, (short)0, acc[t], false, false);
            acc[t] = __builtin_amdgcn_wmma_f32_16x16x4_f32(
                false, aA, false, bL, (short)0, acc[t], false, false);
        }
    }

    // D layout: reg v -> row = v + 8*khalf, col = lane&15 (per 16-col tile)
#pragma unroll
    for (int t = 0; t < 4; ++t) {
        const int col = t * 16 + m;
#pragma unroll
        for (int v = 0; v < 8; ++v) {
            const int row = rowb + v + 8 * khalf;
            float x = acc[t][v];
            OUT[row * FEAT + col] = x > 0.0f ? x : 0.0f;
        }
    }
}

// ------------------------------------------------------- pooled segment sum
__global__ __launch_bounds__(256) void gnn_pool_sum_kernel(
    const float* __restrict__ h,      // [N, 64]
    const int* __restrict__ batch,    // [N]
    float* __restrict__ sums,         // [G, 64]
    float* __restrict__ counts)       // [G]
{
    int idx = blockIdx.x * blockDim.x + threadIdx.x;
    if (idx >= NF) return;
    int node = idx >> 6;
    int f    = idx & 63;
    int g    = batch[node];
    atomicAdd(sums + g * FEAT + f, h[idx]);
    if (f == 0) atomicAdd(counts + g, 1.0f);
}

// ---------------------------------------------------- mean pool + 2 FC head
__global__ __launch_bounds__(64) void gnn_head_kernel(
    const float* __restrict__ sums, const float* __restrict__ counts,
    const float* __restrict__ Wf1, const float* __restrict__ bf1,
    const float* __restrict__ Wf2, const float* __restrict__ bf2,
    float* __restrict__ out)          // [G, 2]
{
    __shared__ float pooled[FEAT];
    __shared__ float t1[FEAT];
    const int g = blockIdx.x;
    const int j = threadIdx.x;

    float c = counts[g];
    c = c > 1.0f ? c : 1.0f;
    pooled[j] = sums[g * FEAT + j] / c;
    __syncthreads();

    float a = bf1[j];
#pragma unroll 8
    for (int k = 0; k < FEAT; ++k) a += pooled[k] * Wf1[k * FEAT + j];
    t1[j] = a;
    __syncthreads();

    if (j < 2) {
        float o = bf2[j];
        for (int k = 0; k < FEAT; ++k) o += t1[k] * Wf2[k * 2 + j];
        out[g * 2 + j] = o;
    }
}

// ---------------------------------------------------------------- launcher
extern "C" void kernel_launch(void* const* d_in, const int* in_sizes, int n_in,
                              void* d_out, int out_size, void* d_ws, size_t ws_size,
                              hipStream_t stream) {
    const float* x      = (const float*)d_in[0];
    const int*   ei     = (const int*)  d_in[1];
    const int*   batch  = (const int*)  d_in[2];
    const float* W1_rel = (const float*)d_in[3];
    const float* W1_root= (const float*)d_in[4];
    const float* b1     = (const float*)d_in[5];
    const float* W2_rel = (const float*)d_in[6];
    const float* W2_root= (const float*)d_in[7];
    const float* b2     = (const float*)d_in[8];
    const float* W3_rel = (const float*)d_in[9];
    const float* W3_root= (const float*)d_in[10];
    const float* b3     = (const float*)d_in[11];
    const float* Wf1    = (const float*)d_in[12];
    const float* bf1    = (const float*)d_in[13];
    const float* Wf2    = (const float*)d_in[14];
    const float* bf2    = (const float*)d_in[15];
    float* out = (float*)d_out;

    float* agg    = (float*)d_ws;          // [N,64]
    float* hA     = agg + NF;              // [N,64] (h1, then reused for h3)
    float* hB     = hA  + NF;              // [N,64] (h2)
    float* sums   = hB  + NF;              // [G,64]
    float* counts = sums + N_GRAPHS * FEAT;// [G]

    const int zeroBlk = 256;
    const int nfBlocks = (NF + zeroBlk - 1) / zeroBlk;
    const int scatBlocks = (N_EDGES * 16 + zeroBlk - 1) / zeroBlk;
    const int gemmBlocks = (N_NODES / 16 + 3) / 4;   // 4 waves/block, 16 rows/wave
    const int poolBlocks = (NF + zeroBlk - 1) / zeroBlk;
    const int zsumN = N_GRAPHS * FEAT + N_GRAPHS;
    const int zsumBlocks = (zsumN + zeroBlk - 1) / zeroBlk;

    // ---- layer 1
    gnn_zero_kernel<<<nfBlocks, zeroBlk, 0, stream>>>(agg, NF);
    gnn_edge_scatter_kernel<<<scatBlocks, zeroBlk, 0, stream>>>(ei, x, agg);
    gnn_gemm_relu_kernel<<<gemmBlocks, 128, 0, stream>>>(x, agg, W1_root, W1_rel, b1, hA, N_NODES);
    // ---- layer 2
    gnn_zero_kernel<<<nfBlocks, zeroBlk, 0, stream>>>(agg, NF);
    gnn_edge_scatter_kernel<<<scatBlocks, zeroBlk, 0, stream>>>(ei, hA, agg);
    gnn_gemm_relu_kernel<<<gemmBlocks, 128, 0, stream>>>(hA, agg, W2_root, W2_rel, b2, hB, N_NODES);
    // ---- layer 3
    gnn_zero_kernel<<<nfBlocks, zeroBlk, 0, stream>>>(agg, NF);
    gnn_edge_scatter_kernel<<<scatBlocks, zeroBlk, 0, stream>>>(ei, hB, agg);
    gnn_gemm_relu_kernel<<<gemmBlocks, 128, 0, stream>>>(hB, agg, W3_root, W3_rel, b3, hA, N_NODES);
    // ---- mean pool + head
    gnn_zero_kernel<<<zsumBlocks, zeroBlk, 0, stream>>>(sums, zsumN);
    gnn_pool_sum_kernel<<<poolBlocks, zeroBlk, 0, stream>>>(hA, batch, sums, counts);
    gnn_head_kernel<<<N_GRAPHS, 64, 0, stream>>>(sums, counts, Wf1, bf1, Wf2, bf2, out);
}